// GATDetector_25924422598990
// MI455X (gfx1250) — compile-verified
//
#include <hip/hip_runtime.h>
#include <hip/hip_bf16.h>
#include <cmath>

typedef __attribute__((ext_vector_type(2))) float v2f;
typedef __attribute__((ext_vector_type(8))) float v8f;

#define NEG_SLOPE 0.2f

// ---------------------------------------------------------------- utilities

__global__ void k_fill(float* __restrict__ p, size_t n, float v) {
    size_t i = (size_t)blockIdx.x * blockDim.x + threadIdx.x;
    if (i < n) p[i] = v;
}

// CDNA5 native no-return float atomic max (GLOBAL_ATOMIC_MAX_NUM_F32, op 82)
__device__ __forceinline__ void atomic_max_f32(float* addr, float val) {
    asm volatile("global_atomic_max_num_f32 %0, %1, off scope:SCOPE_DEV"
                 :: "v"(addr), "v"(val) : "memory");
}

__device__ __forceinline__ void edge_sd(const int* __restrict__ ei, int E,
                                        int e, int& s, int& d) {
    if (e < E) { s = ei[e]; d = ei[E + e]; }
    else       { s = e - E; d = e - E; }      // appended self-loops
}

// ------------------------------------------------- layer 1: x@W1 + attn dots

// one node per block; 256 threads compute h1[n, 0..255], then 8 threads
// fold the per-head attention dot products out of LDS.
__global__ __launch_bounds__(256) void k_node1(
    const float* __restrict__ x, const float* __restrict__ W1,
    const float* __restrict__ att_s, const float* __restrict__ att_d,
    float* __restrict__ h1, float* __restrict__ as1, float* __restrict__ ad1) {
    __shared__ float sh[256];
    int n = blockIdx.x, c = threadIdx.x;
    float xv[5];
#pragma unroll
    for (int k = 0; k < 5; ++k) xv[k] = x[(size_t)n * 5 + k];
    float acc = 0.f;
#pragma unroll
    for (int k = 0; k < 5; ++k) acc += xv[k] * W1[k * 256 + c];
    h1[(size_t)n * 256 + c] = acc;
    sh[c] = acc;
    __syncthreads();
    if (threadIdx.x < 8) {
        int h = threadIdx.x & 3;
        const float* att = (threadIdx.x >= 4) ? att_d : att_s;
        float a = 0.f;
#pragma unroll 8
        for (int cc = 0; cc < 64; ++cc) a += sh[h * 64 + cc] * att[h * 64 + cc];
        if (threadIdx.x >= 4) ad1[n * 4 + h] = a; else as1[n * 4 + h] = a;
    }
}

// ----------------------------------------------- layer 1: edge softmax passes

__global__ __launch_bounds__(256) void k_edge_max1(
    const int* __restrict__ ei, int E, const float* __restrict__ as1,
    const float* __restrict__ ad1, float* __restrict__ mx1, int Etot) {
    int e = blockIdx.x * 256 + threadIdx.x;
    if (e >= Etot) return;
    int s, d; edge_sd(ei, E, e, s, d);
#pragma unroll
    for (int h = 0; h < 4; ++h) {
        float v = as1[s * 4 + h] + ad1[d * 4 + h];
        v = v > 0.f ? v : NEG_SLOPE * v;
        atomic_max_f32(&mx1[d * 4 + h], v);
    }
}

__global__ __launch_bounds__(256) void k_edge_sum1(
    const int* __restrict__ ei, int E, const float* __restrict__ as1,
    const float* __restrict__ ad1, const float* __restrict__ mx1,
    float* __restrict__ dn1, float* __restrict__ ex1, int Etot) {
    int e = blockIdx.x * 256 + threadIdx.x;
    if (e >= Etot) return;
    int s, d; edge_sd(ei, E, e, s, d);
#pragma unroll
    for (int h = 0; h < 4; ++h) {
        float v = as1[s * 4 + h] + ad1[d * 4 + h];
        v = v > 0.f ? v : NEG_SLOPE * v;
        float ex = __expf(v - mx1[d * 4 + h]);
        ex1[(size_t)e * 4 + h] = ex;
        atomicAdd(&dn1[d * 4 + h], ex);
    }
}

// one block per edge; thread t covers (head t>>6, channel t&63) == channel t of 256
__global__ __launch_bounds__(256) void k_edge_agg1(
    const int* __restrict__ ei, int E, const float* __restrict__ ex1,
    const float* __restrict__ dn1, const float* __restrict__ h1,
    float* __restrict__ out1) {
    int e = blockIdx.x, t = threadIdx.x, h = t >> 6;
    int s, d; edge_sd(ei, E, e, s, d);
    float alpha = ex1[(size_t)e * 4 + h] / (dn1[d * 4 + h] + 1e-16f);
    float v = alpha * h1[(size_t)s * 256 + t];
    atomicAdd(&out1[(size_t)d * 256 + t], v);
}

__global__ __launch_bounds__(256) void k_bias_relu1(
    float* __restrict__ h, const float* __restrict__ b, size_t n) {
    size_t i = (size_t)blockIdx.x * 256 + threadIdx.x;
    if (i >= n) return;
    float v = h[i] + b[i & 255];
    h[i] = v > 0.f ? v : 0.f;
}

// ------------------------------------------- layer 2 GEMM: [N,256]x[256,64]
// fp32 WMMA 16x16x4. Block = 128 threads = 4 waves; each block owns a
// 16-node tile; wave w computes the 16x16 output tile at columns w*16.
// A tile (16x256) staged in LDS once (padded stride 260 -> conflict-free).

__global__ __launch_bounds__(128) void k_gemm2(
    const float* __restrict__ hin, const float* __restrict__ W2,
    float* __restrict__ hout) {
    __shared__ float sA[16][260];
    int m0 = blockIdx.x * 16;
    for (int i = threadIdx.x; i < 16 * 256; i += 128) {
        int r = i >> 8, c = i & 255;
        sA[r][c] = hin[(size_t)(m0 + r) * 256 + c];
    }
    __syncthreads();
    int wave = threadIdx.x >> 5, lane = threadIdx.x & 31;
    int lm = lane & 15, hi = lane >> 4;
    int n0 = wave * 16;
    v8f acc = {};
#pragma unroll 4
    for (int k0 = 0; k0 < 256; k0 += 4) {
        int ka = k0 + 2 * hi;                 // A 16x4: lanes0-15 K={k0,k0+1}, lanes16-31 K={k0+2,k0+3}
        v2f a, b;
        a.x = sA[lm][ka];
        a.y = sA[lm][ka + 1];
        b.x = W2[(size_t)ka * 64 + n0 + lm];  // B 4x16 mirrored layout
        b.y = W2[(size_t)(ka + 1) * 64 + n0 + lm];
        acc = __builtin_amdgcn_wmma_f32_16x16x4_f32(
            false, a, false, b, (short)0, acc, false, false);
    }
#pragma unroll
    for (int r = 0; r < 8; ++r)               // D: vgpr r -> M = r + 8*hi, N = lm
        hout[(size_t)(m0 + r + 8 * hi) * 64 + n0 + lm] = acc[r];
}

// --------------------------------------------------- layer 2 attention dots

__global__ __launch_bounds__(256) void k_attn2(
    const float* __restrict__ h2, const float* __restrict__ att_s,
    const float* __restrict__ att_d, float* __restrict__ as2,
    float* __restrict__ ad2, int N) {
    int n = blockIdx.x * 256 + threadIdx.x;
    if (n >= N) return;
    float a = 0.f, b = 0.f;
#pragma unroll 8
    for (int c = 0; c < 64; ++c) {
        float v = h2[(size_t)n * 64 + c];
        a += v * att_s[c];
        b += v * att_d[c];
    }
    as2[n] = a; ad2[n] = b;
}

// ----------------------------------------------- layer 2 edge softmax passes

__global__ __launch_bounds__(256) void k_edge_max2(
    const int* __restrict__ ei, int E, const float* __restrict__ as2,
    const float* __restrict__ ad2, float* __restrict__ mx2, int Etot) {
    int e = blockIdx.x * 256 + threadIdx.x;
    if (e >= Etot) return;
    int s, d; edge_sd(ei, E, e, s, d);
    float v = as2[s] + ad2[d];
    v = v > 0.f ? v : NEG_SLOPE * v;
    atomic_max_f32(&mx2[d], v);
}

__global__ __launch_bounds__(256) void k_edge_sum2(
    const int* __restrict__ ei, int E, const float* __restrict__ as2,
    const float* __restrict__ ad2, const float* __restrict__ mx2,
    float* __restrict__ dn2, float* __restrict__ ex2, int Etot) {
    int e = blockIdx.x * 256 + threadIdx.x;
    if (e >= Etot) return;
    int s, d; edge_sd(ei, E, e, s, d);
    float v = as2[s] + ad2[d];
    v = v > 0.f ? v : NEG_SLOPE * v;
    float ex = __expf(v - mx2[d]);
    ex2[e] = ex;
    atomicAdd(&dn2[d], ex);
}

// 4 edges per 256-thread block; thread covers (edge, channel)
__global__ __launch_bounds__(256) void k_edge_agg2(
    const int* __restrict__ ei, int E, const float* __restrict__ ex2,
    const float* __restrict__ dn2, const float* __restrict__ h2,
    float* __restrict__ out2, int Etot) {
    int e = blockIdx.x * 4 + (threadIdx.x >> 6);
    if (e >= Etot) return;
    int c = threadIdx.x & 63;
    int s, d; edge_sd(ei, E, e, s, d);
    float alpha = ex2[e] / (dn2[d] + 1e-16f);
    atomicAdd(&out2[(size_t)d * 64 + c], alpha * h2[(size_t)s * 64 + c]);
}

// -------------------------------- bias+relu+graph mean-pool (LDS staged)

__global__ __launch_bounds__(256) void k_pool(
    const float* __restrict__ out2, const float* __restrict__ b2,
    const int* __restrict__ batch, float* __restrict__ pool,
    float* __restrict__ cnt, int N) {
    __shared__ float sp[64 * 64];
    __shared__ float sc[64];
    for (int i = threadIdx.x; i < 64 * 64; i += 256) sp[i] = 0.f;
    if (threadIdx.x < 64) sc[threadIdx.x] = 0.f;
    __syncthreads();
    int n = blockIdx.x * 256 + threadIdx.x;
    if (n < N) {
        int g = batch[n];
        atomicAdd(&sc[g], 1.f);
#pragma unroll 4
        for (int c = 0; c < 64; ++c) {
            float v = out2[(size_t)n * 64 + c] + b2[c];
            v = v > 0.f ? v : 0.f;
            atomicAdd(&sp[g * 64 + c], v);
        }
    }
    __syncthreads();
    for (int i = threadIdx.x; i < 64 * 64; i += 256) {
        float v = sp[i];
        if (v != 0.f) atomicAdd(&pool[i], v);
    }
    if (threadIdx.x < 64) {
        float v = sc[threadIdx.x];
        if (v != 0.f) atomicAdd(&cnt[threadIdx.x], v);
    }
}

// --------------------------------------------- final tiny linear [64,64]x[64,2]

__global__ void k_final(const float* __restrict__ pool,
                        const float* __restrict__ cnt,
                        const float* __restrict__ linW,
                        const float* __restrict__ linb,
                        float* __restrict__ out) {
    int t = threadIdx.x;
    if (t >= 128) return;
    int g = t >> 1, j = t & 1;
    float cn = cnt[g]; cn = cn > 1.f ? cn : 1.f;
    float acc = 0.f;
#pragma unroll 8
    for (int c = 0; c < 64; ++c)
        acc += (pool[g * 64 + c] / cn) * linW[c * 2 + j];
    out[t] = acc + linb[j];
}

// ------------------------------------------------------------------ launcher

extern "C" void kernel_launch(void* const* d_in, const int* in_sizes, int n_in,
                              void* d_out, int out_size, void* d_ws, size_t ws_size,
                              hipStream_t stream) {
    const float* x    = (const float*)d_in[0];
    const int*   ei   = (const int*)d_in[1];
    const int*   batch= (const int*)d_in[2];
    const float* W1   = (const float*)d_in[3];
    const float* aS1  = (const float*)d_in[4];
    const float* aD1  = (const float*)d_in[5];
    const float* b1   = (const float*)d_in[6];
    const float* W2   = (const float*)d_in[7];
    const float* aS2  = (const float*)d_in[8];
    const float* aD2  = (const float*)d_in[9];
    const float* b2   = (const float*)d_in[10];
    const float* linW = (const float*)d_in[11];
    const float* linb = (const float*)d_in[12];
    float* out = (float*)d_out;

    const int N    = in_sizes[0] / 5;   // 50000
    const int E    = in_sizes[1] / 2;   // 800000
    const int Etot = E + N;             // + self loops

    // workspace carve-up (floats)
    float* ws = (float*)d_ws;
    size_t o = 0;
    float* h1   = ws + o; o += (size_t)N * 256;   // layer-1 features
    float* agg1 = ws + o; o += (size_t)N * 256;   // layer-1 aggregate -> relu'd h
    float* as1  = ws + o; o += (size_t)N * 4;
    float* ad1  = ws + o; o += (size_t)N * 4;
    float* mx1  = ws + o; o += (size_t)N * 4;
    float* dn1  = ws + o; o += (size_t)N * 4;
    float* ex1  = ws + o; o += (size_t)Etot * 4;
    float* as2  = ws + o; o += (size_t)N;
    float* ad2  = ws + o; o += (size_t)N;
    float* mx2  = ws + o; o += (size_t)N;
    float* dn2  = ws + o; o += (size_t)N;
    float* pool = ws + o; o += 64 * 64;
    float* cnt  = ws + o; o += 64;
    float* h2   = h1;     // reuse: h1 dead after edge_agg1
    float* ex2  = ex1;    // reuse: ex1 dead after edge_agg1
    float* out2 = agg1;   // reuse: agg1 dead after gemm2 reads it

    auto cdiv = [](size_t a, size_t b) { return (unsigned)((a + b - 1) / b); };

    // ---- layer 1 ----
    k_fill<<<cdiv((size_t)N * 256, 256), 256, 0, stream>>>(agg1, (size_t)N * 256, 0.f);
    k_fill<<<cdiv((size_t)N * 4, 256), 256, 0, stream>>>(mx1, (size_t)N * 4, -INFINITY);
    k_fill<<<cdiv((size_t)N * 4, 256), 256, 0, stream>>>(dn1, (size_t)N * 4, 0.f);

    k_node1<<<N, 256, 0, stream>>>(x, W1, aS1, aD1, h1, as1, ad1);
    k_edge_max1<<<cdiv(Etot, 256), 256, 0, stream>>>(ei, E, as1, ad1, mx1, Etot);
    k_edge_sum1<<<cdiv(Etot, 256), 256, 0, stream>>>(ei, E, as1, ad1, mx1, dn1, ex1, Etot);
    k_edge_agg1<<<Etot, 256, 0, stream>>>(ei, E, ex1, dn1, h1, agg1);
    k_bias_relu1<<<cdiv((size_t)N * 256, 256), 256, 0, stream>>>(agg1, b1, (size_t)N * 256);

    // ---- layer 2 ----
    k_gemm2<<<N / 16, 128, 0, stream>>>(agg1, W2, h2);
    k_attn2<<<cdiv(N, 256), 256, 0, stream>>>(h2, aS2, aD2, as2, ad2, N);

    k_fill<<<cdiv((size_t)N * 64, 256), 256, 0, stream>>>(out2, (size_t)N * 64, 0.f);
    k_fill<<<cdiv(N, 256), 256, 0, stream>>>(mx2, N, -INFINITY);
    k_fill<<<cdiv(N, 256), 256, 0, stream>>>(dn2, N, 0.f);
    k_fill<<<1, 256, 0, stream>>>(pool, 64 * 64 + 64, 0.f);  // pool + cnt contiguous

    k_edge_max2<<<cdiv(Etot, 256), 256, 0, stream>>>(ei, E, as2, ad2, mx2, Etot);
    k_edge_sum2<<<cdiv(Etot, 256), 256, 0, stream>>>(ei, E, as2, ad2, mx2, dn2, ex2, Etot);
    k_edge_agg2<<<cdiv(Etot, 4), 256, 0, stream>>>(ei, E, ex2, dn2, h2, out2, Etot);

    // ---- pool + classifier ----
    k_pool<<<cdiv(N, 256), 256, 0, stream>>>(out2, b2, batch, pool, cnt, N);
    k_final<<<1, 128, 0, stream>>>(pool, cnt, linW, linb, out);
}